// TFRToSeqFTPlaneConvCollapse_352187318340
// MI455X (gfx1250) — compile-verified
//
#include <hip/hip_runtime.h>

typedef __attribute__((ext_vector_type(2))) float v2f;
typedef __attribute__((ext_vector_type(8))) float v8f;

#define C_DIM 64
#define F_DIM 256
#define T_DIM 1024
#define CH_STRIDE (F_DIM * T_DIM)   // 262144 elements between channel planes

// ---------------------------------------------------------------------------
// Pass 1: channel sum via V_WMMA_F32_16X16X4_F32 (D = ones(16x4) * B(4x16) + C)
// One wave sums all 64 channels for 16 consecutive t-values of one (b,f) row.
// Block = 8 waves -> covers 128 t. Grid = (B*F rows) * 8 = 16384 blocks.
// ---------------------------------------------------------------------------
__global__ __launch_bounds__(256) void chan_sum_wmma(const float* __restrict__ x,
                                                     float* __restrict__ xs) {
    const int lane = threadIdx.x & 31;
    const int wave = threadIdx.x >> 5;
    const int row  = blockIdx.x >> 3;                       // b*F + f, 0..2047
    const int t0   = ((blockIdx.x & 7) << 7) + (wave << 4); // 0..1023 step 16

    const int b = row >> 8;        // /256
    const int f = row & 255;
    const int n = lane & 15;       // N index within the 16-wide t tile

    // Lanes 0-15 carry channels {c, c+1}; lanes 16-31 carry {c+2, c+3}.
    // (With A == all-ones, D = sum_k B[k][n] regardless of K<->VGPR mapping,
    //  so this is robust to the B operand's K striping.)
    const long chanOfs = (lane < 16) ? 0L : 2L * CH_STRIDE;
    const float* q = x + (long)b * (C_DIM * (long)CH_STRIDE)
                       + (long)f * T_DIM + t0 + n + chanOfs;

    v2f ones; ones.x = 1.0f; ones.y = 1.0f;
    v8f acc = {};

#pragma unroll
    for (int i = 0; i < 16; ++i) {          // 16 steps x 4 channels = 64
        v2f bmat;
        bmat.x = q[0];                       // channel c (+0 / +2)
        bmat.y = q[CH_STRIDE];               // channel c+1 (+1 / +3), 1 MB imm offset
        acc = __builtin_amdgcn_wmma_f32_16x16x4_f32(
                  /*neg_a=*/false, ones,
                  /*neg_b=*/false, bmat,
                  /*c_mod=*/(short)0, acc,
                  /*reuse_a=*/false, /*reuse_b=*/false);
        q += 4L * CH_STRIDE;
    }

    // Every lane's acc[0] holds the channel sum for t0 + (lane & 15);
    // lanes 16-31 are duplicates, so only the low half stores.
    if (lane < 16) {
        xs[(long)row * T_DIM + t0 + n] = acc[0];
    }
}

// ---------------------------------------------------------------------------
// Pass 2: 3x3 cross-correlation (zero-padded) + bias + transpose to (B,T,F).
// xs is 8.4 MB -> L2 resident. 32x32 output tile per block, 34x34 halo in LDS.
// ---------------------------------------------------------------------------
__global__ __launch_bounds__(256) void conv3x3_bias_transpose(
    const float* __restrict__ xs, const float* __restrict__ w9,
    const float* __restrict__ bias, float* __restrict__ out) {
    __shared__ float tile[34][35];          // stride 35 -> no bank conflicts

    const int tid   = threadIdx.x;
    const int tTile = blockIdx.x & 31;       // 32 t tiles
    const int fTile = (blockIdx.x >> 5) & 7; // 8 f tiles
    const int b     = blockIdx.x >> 8;       // 8 batches
    const int f0 = fTile << 5;
    const int t0 = tTile << 5;

    for (int i = tid; i < 34 * 34; i += 256) {
        const int fr = i / 34;
        const int tc = i - fr * 34;
        const int gf = f0 - 1 + fr;
        const int gt = t0 - 1 + tc;
        float v = 0.0f;
        if ((unsigned)gf < F_DIM && (unsigned)gt < T_DIM)
            v = xs[((long)b * F_DIM + gf) * T_DIM + gt];
        tile[fr][tc] = v;
    }
    __syncthreads();

    const float w00 = w9[0], w01 = w9[1], w02 = w9[2];
    const float w10 = w9[3], w11 = w9[4], w12 = w9[5];
    const float w20 = w9[6], w21 = w9[7], w22 = w9[8];
    const float bs  = bias[0];

    const int fl  = tid & 31;    // consecutive lanes -> consecutive f (coalesced stores)
    const int tl0 = tid >> 5;

#pragma unroll
    for (int k = 0; k < 4; ++k) {
        const int tl = tl0 + (k << 3);
        const float s =
            w00 * tile[fl + 0][tl + 0] + w01 * tile[fl + 0][tl + 1] + w02 * tile[fl + 0][tl + 2] +
            w10 * tile[fl + 1][tl + 0] + w11 * tile[fl + 1][tl + 1] + w12 * tile[fl + 1][tl + 2] +
            w20 * tile[fl + 2][tl + 0] + w21 * tile[fl + 2][tl + 1] + w22 * tile[fl + 2][tl + 2];
        out[((long)b * T_DIM + (t0 + tl)) * F_DIM + (f0 + fl)] = s + bs;
    }
}

extern "C" void kernel_launch(void* const* d_in, const int* in_sizes, int n_in,
                              void* d_out, int out_size, void* d_ws, size_t ws_size,
                              hipStream_t stream) {
    const float* x    = (const float*)d_in[0];   // (8,64,256,1024) fp32
    const float* k9   = (const float*)d_in[1];   // (3,3) fp32
    const float* bias = (const float*)d_in[2];   // (1,)  fp32
    float* out = (float*)d_out;                  // (8,1024,256) fp32
    float* xs  = (float*)d_ws;                   // needs 8*256*1024*4 = 8 MiB scratch

    chan_sum_wmma<<<16384, 256, 0, stream>>>(x, xs);
    conv3x3_bias_transpose<<<2048, 256, 0, stream>>>(xs, k9, bias, out);
}